// FilterDistance_52776558133736
// MI455X (gfx1250) — compile-verified
//
#include <hip/hip_runtime.h>
#include <stdint.h>

#ifndef __has_builtin
#define __has_builtin(x) 0
#endif

#define B_DIM   8
#define N_DIM   1024
#define M_DIM   2048
#define TH2     49.0f              // 7.0^2 : compare squared distance
#define NSTRIPS (B_DIM * (N_DIM / 16))   // 512 blocks, each = 16 agents x all M
#define TILES_M (M_DIM / 16)             // 128 WMMA tiles per strip

typedef __attribute__((ext_vector_type(2))) float v2f;
typedef __attribute__((ext_vector_type(4))) float v4f;
typedef __attribute__((ext_vector_type(8))) float v8f;

// Exact pointee type of the async-LDS builtin params (per hipcc diagnostic):
// "__attribute__((__vector_size__(4 * sizeof(int)))) int __device__ *"
typedef int v4i __attribute__((__vector_size__(4 * sizeof(int))));
typedef __attribute__((address_space(1))) v4i* as1_v4i_p;  // global
typedef __attribute__((address_space(3))) v4i* as3_v4i_p;  // LDS

#if __has_builtin(__builtin_amdgcn_global_load_async_to_lds_b128)
#define HAVE_ASYNC_LDS 1
#endif

__device__ __forceinline__ void wait_async0() {
#if __has_builtin(__builtin_amdgcn_s_wait_asynccnt)
  __builtin_amdgcn_s_wait_asynccnt(0);
#else
  asm volatile("s_wait_asynccnt 0" ::: "memory");
#endif
}

__device__ __forceinline__ unsigned ballot32(bool p) {
#if __has_builtin(__builtin_amdgcn_ballot_w32)
  return __builtin_amdgcn_ballot_w32(p);
#else
  return (unsigned)__ballot(p);
#endif
}

// Build the 16 x 2048 validity bitmask for one strip (batch b, agents n0..n0+15)
// using WMMA f32 16x16x4:
//   A row i  = [x_i, y_i, x_i^2+y_i^2, 1]
//   B col j  = [-2u_j, -2v_j, 1, u_j^2+v_j^2]
//   (A.B)_ij = (x_i-u_j)^2 + (y_i-v_j)^2
// Mask layout: mask32[t*8 + r] = { lo u16: row r cols [16t,16t+16),
//                                  hi u16: row r+8 same cols }  (r = 0..7)
// This matches the WMMA D layout (VGPR r = rows r / r+8 across lane halves),
// so each accumulator VGPR maps to exactly one ballot -> one u32 word.
__device__ __forceinline__ void build_strip_mask(
    const float* __restrict__ agt, const float* __restrict__ ctx,
    int b, int n0, float* ctxLds, float* agtLds, unsigned* mask32)
{
  const int tid  = threadIdx.x;
  const int lane = tid & 31;
  const int wave = tid >> 5;
  const int l15  = lane & 15;

  // Stage this batch's 2048 (u,v) ctx centers (16KB) into LDS.
  const float* gctx = ctx + (size_t)b * M_DIM * 2;
  {
    v4f*       lds4 = (v4f*)ctxLds;
    const v4f* g4   = (const v4f*)gctx;
#ifdef HAVE_ASYNC_LDS
#pragma unroll
    for (int k = 0; k < 4; ++k) {
      int idx = tid + k * 256;
      __builtin_amdgcn_global_load_async_to_lds_b128(
          (as1_v4i_p)(g4 + idx),
          (as3_v4i_p)(lds4 + idx),
          0, 0);
    }
#else
#pragma unroll
    for (int k = 0; k < 4; ++k) { int idx = tid + k * 256; lds4[idx] = g4[idx]; }
#endif
  }
  if (tid < 16) {
    const float* ga = agt + ((size_t)b * N_DIM + n0 + tid) * 2;
    agtLds[2 * tid]     = ga[0];
    agtLds[2 * tid + 1] = ga[1];
  }
#ifdef HAVE_ASYNC_LDS
  wait_async0();
#endif
  __syncthreads();

  // A fragment (16x4 f32): lanes 0-15 hold K=0,1 ; lanes 16-31 hold K=2,3.
  float ax = agtLds[2 * l15], ay = agtLds[2 * l15 + 1];
  v2f afrag;
  afrag.x = (lane < 16) ? ax : (ax * ax + ay * ay);
  afrag.y = (lane < 16) ? ay : 1.0f;

  for (int t = wave; t < TILES_M; t += 8) {
    const int mb = t << 4;
    // B fragment (4x16 f32): lanes 0-15 hold K=0,1 ; lanes 16-31 hold K=2,3.
    float u = ctxLds[2 * (mb + l15)], v = ctxLds[2 * (mb + l15) + 1];
    v2f bfrag;
    bfrag.x = (lane < 16) ? (-2.0f * u) : 1.0f;
    bfrag.y = (lane < 16) ? (-2.0f * v) : (u * u + v * v);

    v8f acc = {};
    acc = __builtin_amdgcn_wmma_f32_16x16x4_f32(
        false, afrag, false, bfrag, (short)0, acc, false, false);

    // All 8 ballots first (v_cmp -> SGPR, full EXEC, branch-free) ...
    unsigned bal[8];
#pragma unroll
    for (int r = 0; r < 8; ++r) bal[r] = ballot32(acc[r] <= TH2);
    // ... then one single-lane region with 8 consecutive b32 stores
    // (mergeable into ds_store_b128 pairs).
    if (lane == 0) {
      unsigned* w = mask32 + (t << 3);
#pragma unroll
      for (int r = 0; r < 8; ++r) w[r] = bal[r];
    }
  }
  __syncthreads();
}

// Pass 0: fill the entire padded output (335 MB) with reference defaults.
// Bandwidth-bound floor of the whole problem; use non-temporal B128 stores.
__global__ void __launch_bounds__(256)
fd_fill_kernel(const float* __restrict__ agt, const float* __restrict__ ctx,
               float* __restrict__ out)
{
  const size_t K      = (size_t)B_DIM * N_DIM * M_DIM;
  const size_t total4 = (5 * K) / 4;   // whole output in float4 units
  const size_t d4     = (2 * K) / 4;   // d-region in float4 units
  float d0x = agt[0] - ctx[0];
  float d0y = agt[1] - ctx[1];
  v4f pat;  pat.x = d0x; pat.y = d0y; pat.z = d0x; pat.w = d0y;
  v4f zero = {};
  v4f* out4 = (v4f*)out;
  size_t stride = (size_t)gridDim.x * blockDim.x;
  for (size_t i = (size_t)blockIdx.x * blockDim.x + threadIdx.x; i < total4; i += stride) {
    v4f val = (i < d4) ? pat : zero;
    __builtin_nontemporal_store(val, out4 + i);
  }
}

// Pass 1: per-strip valid count.
__global__ void __launch_bounds__(256)
fd_count_kernel(const float* __restrict__ agt, const float* __restrict__ ctx,
                unsigned* __restrict__ counts)
{
  __shared__ float ctxLds[M_DIM * 2];
  __shared__ float agtLds[32];
  __shared__ unsigned mask32[TILES_M * 8];
  __shared__ unsigned partial[256];
  const int blk = blockIdx.x;
  const int b   = blk >> 6;
  const int n0  = (blk & 63) << 4;
  build_strip_mask(agt, ctx, b, n0, ctxLds, agtLds, mask32);

  const int tid = threadIdx.x;
  unsigned s = 0;
  for (int i = tid; i < TILES_M * 8; i += 256) s += __popc(mask32[i]);
  partial[tid] = s;
  __syncthreads();
  for (int off = 128; off > 0; off >>= 1) {
    if (tid < off) partial[tid] += partial[tid + off];
    __syncthreads();
  }
  if (tid == 0) counts[blk] = partial[0];
}

// Pass 2: exclusive scan of 512 strip counts (strips are contiguous in flat
// (b,n,m) order, so scanned offsets give the reference-ordered bases).
__global__ void __launch_bounds__(NSTRIPS)
fd_scan_kernel(const unsigned* __restrict__ counts, unsigned* __restrict__ offsets)
{
  __shared__ unsigned sc[NSTRIPS];
  const int tid = threadIdx.x;
  unsigned c = counts[tid];
  sc[tid] = c;
  __syncthreads();
  for (int off = 1; off < NSTRIPS; off <<= 1) {
    unsigned v = (tid >= off) ? sc[tid - off] : 0u;
    __syncthreads();
    sc[tid] += v;
    __syncthreads();
  }
  offsets[tid] = sc[tid] - c;   // exclusive
}

// Pass 3: rebuild masks, rank every valid pair (row bases + intra-row tile
// prefixes + in-word popcount), write compacted outputs.
__global__ void __launch_bounds__(256)
fd_write_kernel(const float* __restrict__ agt, const float* __restrict__ ctx,
                const unsigned* __restrict__ offsets, float* __restrict__ out)
{
  __shared__ float ctxLds[M_DIM * 2];
  __shared__ float agtLds[32];
  __shared__ unsigned mask32[TILES_M * 8];
  __shared__ unsigned short prefix16[16][TILES_M];
  __shared__ unsigned rowcnt[16];
  __shared__ unsigned rowbase[16];
  const int blk = blockIdx.x;
  const int b   = blk >> 6;
  const int n0  = (blk & 63) << 4;
  build_strip_mask(agt, ctx, b, n0, ctxLds, agtLds, mask32);

  const int tid = threadIdx.x;
  if (tid < 16) {                       // per-row running prefix over 128 tiles
    const int sub = tid & 7;            // word slot within tile
    const int hi  = tid >> 3;           // 0: low u16 (rows 0-7), 1: high u16
    unsigned run = 0;
    for (int t = 0; t < TILES_M; ++t) {
      unsigned full = mask32[(t << 3) + sub];
      unsigned w = hi ? (full >> 16) : (full & 0xFFFFu);
      prefix16[tid][t] = (unsigned short)run;
      run += __popc(w);
    }
    rowcnt[tid] = run;
  }
  __syncthreads();
  if (tid == 0) {                       // row bases within strip (flat order)
    unsigned run = 0;
    for (int n = 0; n < 16; ++n) { rowbase[n] = run; run += rowcnt[n]; }
  }
  __syncthreads();

  const size_t K = (size_t)B_DIM * N_DIM * M_DIM;
  float* out_d     = out;
  float* out_hi    = out + 2 * K;
  float* out_wi    = out_hi + K;
  float* out_valid = out_wi + K;
  const unsigned base = offsets[blk];

  for (int e = tid; e < 16 * M_DIM; e += 256) {
    const int n = e >> 11;
    const int m = e & (M_DIM - 1);
    const unsigned full   = mask32[((m >> 4) << 3) + (n & 7)];
    const unsigned w      = (n >= 8) ? (full >> 16) : (full & 0xFFFFu);
    const unsigned bitpos = (unsigned)(m & 15);
    if ((w >> bitpos) & 1u) {
      const unsigned rank = base + rowbase[n] + (unsigned)prefix16[n][m >> 4]
                          + (unsigned)__popc(w & ((1u << bitpos) - 1u));
      const float axv = agtLds[2 * n], ayv = agtLds[2 * n + 1];
      const float u = ctxLds[2 * m],   v = ctxLds[2 * m + 1];
      v2f dv; dv.x = axv - u; dv.y = ayv - v;
      ((v2f*)out_d)[rank] = dv;
      out_hi[rank]    = (float)(b * N_DIM + n0 + n);
      out_wi[rank]    = (float)(b * M_DIM + m);
      out_valid[rank] = 1.0f;
    }
  }
}

extern "C" void kernel_launch(void* const* d_in, const int* in_sizes, int n_in,
                              void* d_out, int out_size, void* d_ws, size_t ws_size,
                              hipStream_t stream)
{
  (void)in_sizes; (void)n_in; (void)out_size; (void)ws_size;
  // dict order: agt_idcs (unused), agt_ctrs, ctx_idcs (unused), ctx_ctrs
  const float* agt = (const float*)d_in[1];
  const float* ctx = (const float*)d_in[3];
  float* out = (float*)d_out;
  unsigned* counts  = (unsigned*)d_ws;        // 512 u32
  unsigned* offsets = counts + NSTRIPS;       // 512 u32

  fd_fill_kernel <<<8192, 256, 0, stream>>>(agt, ctx, out);
  fd_count_kernel<<<NSTRIPS, 256, 0, stream>>>(agt, ctx, counts);
  fd_scan_kernel <<<1, NSTRIPS, 0, stream>>>(counts, offsets);
  fd_write_kernel<<<NSTRIPS, 256, 0, stream>>>(agt, ctx, offsets, out);
}